// Arcface_Head_26817775796583
// MI455X (gfx1250) — compile-verified
//
#include <hip/hip_runtime.h>
#include <math.h>

typedef __attribute__((ext_vector_type(2))) float v2f;
typedef __attribute__((ext_vector_type(8))) float v8f;

#define BATCH   512
#define EMBED   512
#define NCLS    100000
#define NTILE   64           // classes per block
#define NBLK    1563         // ceil(NCLS / NTILE)
#define MCHUNK  128          // batch rows per accumulation chunk
#define KCHUNK  64           // K staged in LDS per stage
#define PAD     68           // LDS row stride (floats): 16B-aligned, conflict-free

#define S_SCALE 64.0f
#define COS_M   0.8775825618903728f
#define SIN_M   0.4794255386042030f
#define TH_C    (-0.8775825618903728f)   // cos(pi - 0.5)
#define MM_C    0.2397127693021015f      // sin(pi - 0.5) * 0.5
#define NEG_BIG (-3.402823466e38f)

__device__ __forceinline__ void merge_ms(float& M, float& S, float m2, float s2) {
    float Mn = fmaxf(M, m2);
    S = __expf(M - Mn) * S + __expf(m2 - Mn) * s2;
    M = Mn;
}

// Low 32 bits of a generic pointer to LDS = LDS byte offset.
__device__ __forceinline__ unsigned lds_off32(const void* p) {
    return (unsigned)(size_t)p;
}

// GLOBAL_LOAD_ASYNC_TO_LDS_B128, GVS mode: uniform SGPR base + per-lane i32 byte offset.
// Tracked by ASYNCcnt; lands directly in LDS without touching VGPRs.
__device__ __forceinline__ void async_copy_b128(unsigned lds_dst, int gbyte_off,
                                                const void* sbase) {
    asm volatile("global_load_async_to_lds_b128 %0, %1, %2"
                 :: "v"(lds_dst), "v"(gbyte_off), "s"(sbase)
                 : "memory");
}

__global__ __launch_bounds__(256) void arcface_main(
    const float* __restrict__ inp, const int* __restrict__ label,
    const float* __restrict__ wgt, float* __restrict__ out,
    float* __restrict__ pmax, float* __restrict__ psum)
{
    __shared__ float lds_a[2][MCHUNK * PAD];    // input chunk:  128 rows x 64 K, x2 buffers
    __shared__ float lds_w[2][NTILE * PAD];     // weight strip:  64 cols x 64 K, x2 buffers
    __shared__ int   lds_label[BATCH];
    __shared__ float lds_sumsq[NTILE];
    __shared__ float lds_invn[NTILE];
    __shared__ float lds_red[MCHUNK][2][2];     // [row][n_wave][{max,sum}]

    const int tid    = threadIdx.x;
    const int lane   = tid & 31;
    const int wave   = tid >> 5;
    const int m_wave = wave >> 1;   // 0..3 -> 32 rows each
    const int n_wave = wave & 1;    // 0..1 -> 32 cols each
    const int la     = lane & 15;
    const int lg     = lane >> 4;   // selects K pair within fragment
    const int nblk   = blockIdx.x;

    lds_label[tid]       = label[tid];
    lds_label[tid + 256] = label[tid + 256];
    if (tid < NTILE) lds_sumsq[tid] = 0.0f;

    // per-thread staging assignment: one class column segment + two input rows
    const int col = tid >> 2;            // 0..63
    const int kq  = (tid & 3) * 16;      // 16-float K segment within chunk
    int cg = nblk * NTILE + col;
    if (cg >= NCLS) cg = NCLS - 1;       // clamp OOB column loads (results discarded)
    const int wbyte0 = (int)(((size_t)cg * EMBED + kq) * sizeof(float));

    // Issue one stage: 12 async b128 per wave (4 weight + 8 input) into buffer `buf`.
    auto issue_stage = [&](int buf, int m_chunk, int k_blk) {
        const int kb = k_blk * KCHUNK * (int)sizeof(float);
        unsigned wl = lds_off32(&lds_w[buf][col * PAD + kq]);
        #pragma unroll
        for (int i = 0; i < 4; ++i)
            async_copy_b128(wl + 16u * i, wbyte0 + kb + 16 * i, wgt);
        #pragma unroll
        for (int rr = col; rr < MCHUNK; rr += 64) {
            unsigned al = lds_off32(&lds_a[buf][rr * PAD + kq]);
            const int ab = (int)((((size_t)(m_chunk * MCHUNK + rr)) * EMBED + kq)
                                 * sizeof(float)) + kb;
            #pragma unroll
            for (int i = 0; i < 4; ++i)
                async_copy_b128(al + 16u * i, ab + 16 * i, inp);
        }
    };

    __syncthreads();                 // sumsq init visible before first atomicAdd
    issue_stage(0, 0, 0);            // prologue: fill buffer 0

    int buf = 0;
    float ssq = 0.0f;

    for (int m_chunk = 0; m_chunk < BATCH / MCHUNK; ++m_chunk) {
        v8f acc[2][2];
        #pragma unroll
        for (int i = 0; i < 2; ++i)
            #pragma unroll
            for (int j = 0; j < 2; ++j)
                #pragma unroll
                for (int r = 0; r < 8; ++r) acc[i][j][r] = 0.0f;

        for (int k_blk = 0; k_blk < EMBED / KCHUNK; ++k_blk) {
            const int stage = m_chunk * (EMBED / KCHUNK) + k_blk;
            if (stage < (BATCH / MCHUNK) * (EMBED / KCHUNK) - 1) {
                // prefetch next stage into the other buffer (overlaps with compute below)
                const int nm = (k_blk == EMBED / KCHUNK - 1) ? m_chunk + 1 : m_chunk;
                const int nk = (k_blk == EMBED / KCHUNK - 1) ? 0 : k_blk + 1;
                issue_stage(buf ^ 1, nm, nk);
                // async loads retire in order: <=12 outstanding => current stage landed
                asm volatile("s_wait_asynccnt 12" ::: "memory");
            } else {
                asm volatile("s_wait_asynccnt 0" ::: "memory");
            }
            __syncthreads();         // current buffer visible to all waves

            // fused row-norm: read back this thread's staged weight segment (chunk 0 only)
            if (m_chunk == 0) {
                const float4* wv = (const float4*)&lds_w[buf][col * PAD + kq];
                #pragma unroll
                for (int i = 0; i < 4; ++i) {
                    float4 v = wv[i];
                    ssq += v.x * v.x + v.y * v.y + v.z * v.z + v.w * v.w;
                }
            }

            // ---- fp32 WMMA inner loop: 2x2 fragments of 16x16, K step 4 ----
            const float* aBase = &lds_a[buf][(m_wave * 32 + la) * PAD];
            const float* wBase = &lds_w[buf][(n_wave * 32 + la) * PAD];
            #pragma unroll 8
            for (int k = 0; k < KCHUNK; k += 4) {
                const int ko = k + 2 * lg;                 // ISA 16x4 A/B fragment layout
                v2f a0 = *(const v2f*)(aBase + ko);        // ds_load_b64
                v2f a1 = *(const v2f*)(aBase + 16 * PAD + ko);
                v2f b0 = *(const v2f*)(wBase + ko);
                v2f b1 = *(const v2f*)(wBase + 16 * PAD + ko);
                acc[0][0] = __builtin_amdgcn_wmma_f32_16x16x4_f32(false, a0, false, b0,
                                (short)0, acc[0][0], false, false);
                acc[0][1] = __builtin_amdgcn_wmma_f32_16x16x4_f32(false, a0, false, b1,
                                (short)0, acc[0][1], false, false);
                acc[1][0] = __builtin_amdgcn_wmma_f32_16x16x4_f32(false, a1, false, b0,
                                (short)0, acc[1][0], false, false);
                acc[1][1] = __builtin_amdgcn_wmma_f32_16x16x4_f32(false, a1, false, b1,
                                (short)0, acc[1][1], false, false);
            }
            __syncthreads();         // all reads of `buf` done: safe target for stage+2
            buf ^= 1;
        }

        // ---- finalize inverse norms once (after all K of first M chunk) ----
        if (m_chunk == 0) {
            atomicAdd(&lds_sumsq[col], ssq);        // ds_add_f32, 4 partials per column
            __syncthreads();
            if (tid < NTILE)
                lds_invn[tid] = 1.0f / fmaxf(sqrtf(lds_sumsq[tid]), 1e-12f);
            __syncthreads();
        }

        // ---- epilogue: arcface transform + logit store ----
        float outv[2][2][8];
        #pragma unroll
        for (int mf = 0; mf < 2; ++mf)
            #pragma unroll
            for (int nf = 0; nf < 2; ++nf) {
                const int n_loc  = n_wave * 32 + nf * 16 + la;
                const int n_glob = nblk * NTILE + n_loc;
                const float invn = lds_invn[n_loc];
                #pragma unroll
                for (int r = 0; r < 8; ++r) {
                    const int m_glob = m_chunk * MCHUNK + m_wave * 32 + mf * 16 + r + 8 * lg;
                    float c    = acc[mf][nf][r] * invn;
                    float sine = sqrtf(fmaxf(1.0f - c * c, 0.0f));
                    float phi  = c * COS_M - sine * SIN_M;
                    phi = (c > TH_C) ? phi : (c - MM_C);
                    float o = ((lds_label[m_glob] == n_glob) ? phi : c) * S_SCALE;
                    if (n_glob < NCLS) out[(size_t)m_glob * NCLS + n_glob] = o;
                    else               o = NEG_BIG;
                    outv[mf][nf][r] = o;
                }
            }

        // ---- per-row (max, sum exp) over this wave's 32 columns ----
        #pragma unroll
        for (int mf = 0; mf < 2; ++mf)
            #pragma unroll
            for (int r = 0; r < 8; ++r) {
                float v0 = outv[mf][0][r], v1 = outv[mf][1][r];
                float mx = fmaxf(v0, v1);
                #pragma unroll
                for (int off = 1; off < 16; off <<= 1)
                    mx = fmaxf(mx, __shfl_xor(mx, off, 32));   // stays within 16-lane half
                float sm = __expf(v0 - mx) + __expf(v1 - mx);
                #pragma unroll
                for (int off = 1; off < 16; off <<= 1)
                    sm += __shfl_xor(sm, off, 32);
                const int rr = m_wave * 32 + mf * 16 + r;
                if (lane == 0)       { lds_red[rr][n_wave][0]     = mx; lds_red[rr][n_wave][1]     = sm; }
                else if (lane == 16) { lds_red[rr + 8][n_wave][0] = mx; lds_red[rr + 8][n_wave][1] = sm; }
            }
        __syncthreads();
        if (tid < MCHUNK) {
            float M = lds_red[tid][0][0], S = lds_red[tid][0][1];
            merge_ms(M, S, lds_red[tid][1][0], lds_red[tid][1][1]);
            const int row = m_chunk * MCHUNK + tid;
            pmax[(size_t)row * NBLK + nblk] = M;
            psum[(size_t)row * NBLK + nblk] = S;
        }
        __syncthreads();
    }
}

__global__ __launch_bounds__(256) void arcface_logsumexp(
    const float* __restrict__ pmax, const float* __restrict__ psum,
    const float* __restrict__ out, const int* __restrict__ label,
    float* __restrict__ row_loss)
{
    __shared__ float sM[256], sS[256];
    const int row = blockIdx.x;
    const int t   = threadIdx.x;
    float M = NEG_BIG, S = 0.0f;
    const float* pm = pmax + (size_t)row * NBLK;
    const float* ps = psum + (size_t)row * NBLK;
    for (int i = t; i < NBLK; i += 256) merge_ms(M, S, pm[i], ps[i]);
    sM[t] = M; sS[t] = S;
    __syncthreads();
    for (int off = 128; off > 0; off >>= 1) {
        if (t < off) {
            float Ma = sM[t], Sa = sS[t];
            merge_ms(Ma, Sa, sM[t + off], sS[t + off]);
            sM[t] = Ma; sS[t] = Sa;
        }
        __syncthreads();
    }
    if (t == 0) {
        float logZ = sM[0] + logf(sS[0]);
        int lab = label[row];
        float v = out[(size_t)row * NCLS + lab];
        row_loss[row] = logZ - v;     // -log p(label)
    }
}

__global__ __launch_bounds__(256) void arcface_finalize(
    const float* __restrict__ row_loss, float* __restrict__ out)
{
    __shared__ float s[256];
    const int t = threadIdx.x;
    s[t] = row_loss[t] + row_loss[t + 256];
    __syncthreads();
    for (int off = 128; off > 0; off >>= 1) {
        if (t < off) s[t] += s[t + off];
        __syncthreads();
    }
    if (t == 0) out[(size_t)BATCH * NCLS] = s[0] / (float)BATCH;
}

extern "C" void kernel_launch(void* const* d_in, const int* in_sizes, int n_in,
                              void* d_out, int out_size, void* d_ws, size_t ws_size,
                              hipStream_t stream) {
    const float* inp   = (const float*)d_in[0];
    const int*   label = (const int*)d_in[1];
    const float* wgt   = (const float*)d_in[2];
    float* out = (float*)d_out;

    float* ws       = (float*)d_ws;
    float* pmax     = ws;                               // [BATCH][NBLK]
    float* psum     = ws + (size_t)BATCH * NBLK;        // [BATCH][NBLK]
    float* row_loss = psum + (size_t)BATCH * NBLK;      // [BATCH]

    arcface_main<<<NBLK, 256, 0, stream>>>(inp, label, wgt, out, pmax, psum);
    arcface_logsumexp<<<BATCH, 256, 0, stream>>>(pmax, psum, out, label, row_loss);
    arcface_finalize<<<1, 256, 0, stream>>>(row_loss, out);
}